// LogBarrierProjection_56495999811646
// MI455X (gfx1250) — compile-verified
//
#include <hip/hip_runtime.h>
#include <hip/hip_bf16.h>
#include <math.h>

// Problem constants (match reference)
#define B_    32
#define S_    16
#define M_    128
#define N_    256
#define P_    (B_ * S_)      // 512 independent problems
#define KIT   30             // K_ITER
#define LSN   25             // MAX_LS
#define MU_   0.01f
#define EPSR  1e-6f

// Padded LDS row stride for A: (260 mod 64) == 4  ->  bank(m,kk) = kk + 4m mod 64,
// conflict-free across the 16 lanes of each WMMA half-wave group.
#define AST   260
#define A_LDS_BYTES (M_ * AST * 4)

typedef __attribute__((ext_vector_type(2))) float v2f;
typedef __attribute__((ext_vector_type(8))) float v8f;
typedef __attribute__((ext_vector_type(4))) int   v4i;

typedef __attribute__((address_space(1))) v4i* gv4i_p;   // global int4*
typedef __attribute__((address_space(3))) v4i* lv4i_p;   // LDS int4*

__device__ __forceinline__ float wave_sum(float v) {
  // wave32 butterfly reduction
  #pragma unroll
  for (int off = 16; off >= 1; off >>= 1) v += __shfl_xor(v, off);
  return v;
}

// ---------------------------------------------------------------------------
// Workspace layout (floats):
//   ws_x     [P_*N_]
//   ws_grad  [P_*N_]
//   ws_invn  [P_*M_]
//   ws_bw    [P_*M_]
//   ws_bthr  [P_*M_]
//   gmask    [KIT]  (as unsigned, after the float arrays)
// ---------------------------------------------------------------------------

__global__ void lbp_mask_init(unsigned* gmask) {
  if (threadIdx.x < KIT) gmask[threadIdx.x] = 0xFFFFFFFFu;
}

// Preamble: row norms, normalized b, thresholds, interior-point init + repair.
__global__ __launch_bounds__(256) void lbp_init_kernel(
    const float* __restrict__ A, const float* __restrict__ b,
    const float* __restrict__ lower, float* __restrict__ ws) {
  const int p = blockIdx.x;
  const int tid = threadIdx.x;
  const int lane = tid & 31, wave = tid >> 5;

  const float* Ap = A + (size_t)p * (M_ * N_);
  const float* bp = b + (size_t)p * M_;
  const float* lop = lower + (size_t)p * N_;

  float* ws_x    = ws;
  float* ws_invn = ws + 2 * (size_t)P_ * N_;
  float* ws_bw   = ws_invn + (size_t)P_ * M_;
  float* ws_bthr = ws_bw + (size_t)P_ * M_;

  __shared__ float s_low[N_];
  __shared__ float s_ratio[M_], s_asum[M_], s_alow[M_], s_invn[M_], s_bw[M_];
  __shared__ float s_t;
  __shared__ int s_feas;

  s_low[tid] = lop[tid];
  if (tid == 0) s_feas = 1;
  __syncthreads();

  // One pass over A: per-row ||row||^2, sum(row), row.lower
  #pragma unroll 1
  for (int r = 0; r < 16; ++r) {
    const int m = wave * 16 + r;
    const float* row = Ap + (size_t)m * N_;
    float n2 = 0.f, asum = 0.f, alow = 0.f;
    #pragma unroll
    for (int n = lane; n < N_; n += 32) {
      float a = row[n];
      n2 = fmaf(a, a, n2);
      asum += a;
      alow = fmaf(a, s_low[n], alow);
    }
    n2 = wave_sum(n2); asum = wave_sum(asum); alow = wave_sum(alow);
    if (lane == 0) {
      float norm = fmaxf(sqrtf(n2), 1e-12f);
      float invn = 1.0f / norm;
      float bw = bp[m] * invn;
      float Ad = invn * asum;              // A_w @ 1
      float sw = bw - invn * alow;         // slack at lower (normalized)
      float ratio = (Ad > 0.f) ? (sw / fmaxf(Ad, 1e-12f)) : INFINITY;
      s_invn[m] = invn; s_bw[m] = bw; s_asum[m] = asum; s_alow[m] = alow;
      s_ratio[m] = ratio;
      ws_invn[p * M_ + m] = invn;
      ws_bw[p * M_ + m]   = bw;
      ws_bthr[p * M_ + m] = bp[m] - EPSR * norm;   // raw-space threshold
    }
  }
  __syncthreads();

  if (tid == 0) {
    float mn = s_ratio[0];
    for (int m = 1; m < M_; ++m) mn = fminf(mn, s_ratio[m]);
    s_t = fmaxf(0.5f * mn, 2.0f * EPSR);
  }
  __syncthreads();
  const float t = s_t;

  // Feasibility of x = lower + t (closed form: A(lower+t) = alow + t*asum)
  if (tid < M_) {
    float lhs = s_invn[tid] * fmaf(t, s_asum[tid], s_alow[tid]);
    if (!(lhs <= s_bw[tid] - EPSR)) atomicAnd(&s_feas, 0);
  }
  __syncthreads();
  const int feas = s_feas && (t >= EPSR);   // x - lower == t everywhere

  float xv = s_low[tid] + t;
  if (!feas) xv = 0.5f * (fmaxf(xv, 0.f) + s_low[tid]);
  ws_x[p * N_ + tid] = xv;
}

// Per-iteration kernel: stage A in LDS (async), Ax, grad, WMMA line-search
// GEMM, global mask AND.
__global__ __launch_bounds__(256) void lbp_iter_kernel(
    const float* __restrict__ A, const float* __restrict__ x_raw,
    const float* __restrict__ lower, float* __restrict__ ws,
    unsigned* __restrict__ gmask, int iter) {
  const int p = blockIdx.x;
  const int tid = threadIdx.x;
  const int lane = tid & 31, wave = tid >> 5;

  const float* Ap = A + (size_t)p * (M_ * N_);
  float* ws_x    = ws;
  float* ws_grad = ws + (size_t)P_ * N_;
  const float* ws_invn = ws + 2 * (size_t)P_ * N_;
  const float* ws_bw   = ws_invn + (size_t)P_ * M_;
  const float* ws_bthr = ws_bw + (size_t)P_ * M_;

  extern __shared__ float As[];            // [M_ * AST], padded row stride
  __shared__ float xs[N_], gs[N_], vs[M_];
  __shared__ unsigned s_mask;

  xs[tid] = ws_x[p * N_ + tid];
  if (tid == 0) s_mask = (1u << LSN) - 1u;

  // ---- stage A (128x256 = 128KB) into LDS, 16B per lane, row-padded ------
#if __has_builtin(__builtin_amdgcn_global_load_async_to_lds_b128) && \
    __has_builtin(__builtin_amdgcn_s_wait_asynccnt)
  #pragma unroll 4
  for (int i = tid; i < M_ * (N_ / 4); i += 256) {
    const int r = i >> 6;
    const int c = (i & 63) << 2;
    __builtin_amdgcn_global_load_async_to_lds_b128(
        (gv4i_p)(Ap + (size_t)r * N_ + c),
        (lv4i_p)(As + r * AST + c),
        0, 0);
  }
  __builtin_amdgcn_s_wait_asynccnt(0);
#else
  #pragma unroll 4
  for (int i = tid; i < M_ * (N_ / 4); i += 256) {
    const int r = i >> 6;
    const int c = (i & 63) << 2;
    const float4 v = *(const float4*)(Ap + (size_t)r * N_ + c);
    *(float4*)(As + r * AST + c) = v;
  }
#endif
  __syncthreads();

  // ---- pass 1: raw Ax per row -> barrier weights v[m] = MU*invn/slack ----
  #pragma unroll 1
  for (int r = 0; r < 16; ++r) {
    const int m = wave * 16 + r;
    const float* row = As + m * AST;
    float acc = 0.f;
    #pragma unroll
    for (int n = lane; n < N_; n += 32) acc = fmaf(row[n], xs[n], acc);
    acc = wave_sum(acc);
    if (lane == 0) {
      float invn = ws_invn[p * M_ + m];
      float slack = fmaxf(ws_bw[p * M_ + m] - invn * acc, 1e-12f);
      vs[m] = MU_ * invn / slack;
    }
  }
  __syncthreads();

  // ---- pass 2: g_bar (A^T v) + gradient + n-side candidate feasibility ----
  {
    const int n = tid;
    float g = 0.f;
    #pragma unroll 4
    for (int m = 0; m < M_; ++m) g = fmaf(As[m * AST + n], vs[m], g);
    const float xv = xs[n];
    const float lo = lower[p * N_ + n];
    const float grad = (xv - x_raw[p * N_ + n]) + g
                     + MU_ / fmaxf(xv - lo, 1e-12f);
    gs[n] = grad;
    ws_grad[p * N_ + n] = grad;

    unsigned nm = 0u;
    float step = 1.0f;                         // LR * BACKTRACK^l
    #pragma unroll
    for (int l = 0; l < LSN; ++l) {
      float c = xv - step * grad;
      if (c >= lo + EPSR) nm |= (1u << l);
      step *= 0.5f;
    }
    atomicAnd(&s_mask, nm);
  }
  __syncthreads();

  // ---- pass 3: WMMA GEMM  Axc[128x32] = A[128x256] * cand[256x32] ----
  // cand[:,l] = x - 2^{-l} * grad  (cols 25..31 unused)
  {
    const int halfsel = lane >> 4;        // K sub-block select
    const int lidx = lane & 15;           // column within tile / row within tile
    const int l0 = lidx, l1 = 16 + lidx;
    const float st0 = (l0 < LSN) ? exp2f(-(float)l0) : 0.f;
    const float st1 = (l1 < LSN) ? exp2f(-(float)l1) : 0.f;
    const int mrow = wave * 16 + lidx;    // A-operand row for this lane
    const float* arow = As + mrow * AST;  // bank-conflict-free (pad = 4 banks/row)

    v8f acc0 = {}; v8f acc1 = {};
    #pragma unroll 4
    for (int k0 = 0; k0 < N_; k0 += 4) {
      const int kk = k0 + (halfsel << 1);
      v2f a;  a.x = arow[kk]; a.y = arow[kk + 1];
      v2f xv; xv.x = xs[kk];  xv.y = xs[kk + 1];
      v2f gv; gv.x = gs[kk];  gv.y = gs[kk + 1];
      v2f b0; b0.x = fmaf(-st0, gv.x, xv.x); b0.y = fmaf(-st0, gv.y, xv.y);
      v2f b1; b1.x = fmaf(-st1, gv.x, xv.x); b1.y = fmaf(-st1, gv.y, xv.y);
      acc0 = __builtin_amdgcn_wmma_f32_16x16x4_f32(false, a, false, b0,
                                                   (short)0, acc0, false, false);
      acc1 = __builtin_amdgcn_wmma_f32_16x16x4_f32(false, a, false, b1,
                                                   (short)0, acc1, false, false);
    }

    // m-side feasibility: raw Axc <= b - EPS*||row||  (per D layout:
    // VGPR j -> M = 16*wave + j + 8*halfsel ; column l = tile*16 + lidx)
    unsigned clear = 0u;
    #pragma unroll
    for (int j = 0; j < 8; ++j) {
      const int m = wave * 16 + j + halfsel * 8;
      const float bt = ws_bthr[p * M_ + m];
      if (!(acc0[j] <= bt) && l0 < LSN) clear |= (1u << l0);
      if (!(acc1[j] <= bt) && l1 < LSN) clear |= (1u << l1);
    }
    if (clear) atomicAnd(&s_mask, ~clear);
  }
  __syncthreads();

  if (tid == 0) atomicAnd(&gmask[iter], s_mask);   // global acceptance AND
}

// Apply globally-chosen step; optionally emit relu(x) on the last iteration.
__global__ __launch_bounds__(256) void lbp_update_kernel(
    float* __restrict__ ws, const unsigned* __restrict__ gmask, int iter,
    float* __restrict__ out, int write_out) {
  const int i = blockIdx.x * blockDim.x + threadIdx.x;   // [0, P_*N_)
  float* ws_x = ws;
  const float* ws_grad = ws + (size_t)P_ * N_;
  const unsigned gm = gmask[iter] & ((1u << LSN) - 1u);
  float step = 0.f;
  if (gm) {
    const int idx = __ffs(gm) - 1;    // first accepted step (argmax of ok_all)
    step = exp2f(-(float)idx);        // LR * BACKTRACK^idx
  }
  const float xn = ws_x[i] - step * ws_grad[i];
  ws_x[i] = xn;
  if (write_out) out[i] = fmaxf(xn, 0.f);
}

extern "C" void kernel_launch(void* const* d_in, const int* in_sizes, int n_in,
                              void* d_out, int out_size, void* d_ws, size_t ws_size,
                              hipStream_t stream) {
  const float* x_raw = (const float*)d_in[0];
  const float* A     = (const float*)d_in[1];
  const float* b     = (const float*)d_in[2];
  const float* lower = (const float*)d_in[3];
  float* ws = (float*)d_ws;
  unsigned* gmask = (unsigned*)(ws + 2 * (size_t)P_ * N_ + 3 * (size_t)P_ * M_);
  float* out = (float*)d_out;

  lbp_mask_init<<<1, 64, 0, stream>>>(gmask);
  lbp_init_kernel<<<P_, 256, 0, stream>>>(A, b, lower, ws);
  for (int k = 0; k < KIT; ++k) {
    lbp_iter_kernel<<<P_, 256, A_LDS_BYTES, stream>>>(A, x_raw, lower, ws,
                                                      gmask, k);
    lbp_update_kernel<<<(P_ * N_) / 256, 256, 0, stream>>>(
        ws, gmask, k, out, (k == KIT - 1) ? 1 : 0);
  }
  (void)in_sizes; (void)n_in; (void)out_size; (void)ws_size;
}